// LRNLayer_29661044146652
// MI455X (gfx1250) — compile-verified
//
#include <hip/hip_runtime.h>
#include <math.h>

// LRN with global per-channel energy, x: (16,256,256,96) f32.
// Bandwidth-bound: min HBM traffic = 2 reads + 1 write of x = ~1.18 GB -> ~51us @ 23.3 TB/s.
// Reduction pass uses V_WMMA_F32_16X16X4_F32 with A = ones so D columns are exact
// f32 per-channel sums of squares (all D rows identical by construction).

#define ALPHA_C 0.0001f
#define BETA_C  0.75f
#define K_C     2.0f
#define CH      96
#define ROWS    (16 * 256 * 256)        // 1,048,576 spatial positions
#define RBLOCKS 1024
#define ROWS_PER_BLOCK (ROWS / RBLOCKS) // 1024, divisible by 4

typedef float v2f __attribute__((ext_vector_type(2)));
typedef float v4f __attribute__((ext_vector_type(4)));
typedef float v8f __attribute__((ext_vector_type(8)));

__global__ void lrn_zero(float* __restrict__ e) {
    e[threadIdx.x] = 0.0f;
}

// 6 waves per block; wave g owns channel group [16g, 16g+16).
// Per WMMA: B holds x^2 for 4 rows x 16 channels (256 B per wave), A = ones,
// so D[m][n] accumulates sum over all K slices of channel n's squares.
// B f32 4x16 layout: lane l -> column N = l%16; half-waves carry different K rows
// (exact K assignment is irrelevant under summation).
__global__ __launch_bounds__(192) void lrn_reduce(const float* __restrict__ x,
                                                  float* __restrict__ e) {
    const int lane = threadIdx.x & 31;
    const int grp  = threadIdx.x >> 5;   // 0..5 channel group
    const int n    = lane & 15;          // column = channel-in-group
    const int h    = lane >> 4;          // half-wave -> K rows {2h, 2h+1}
    const int ch   = grp * 16 + n;

    const long long base = (long long)blockIdx.x * ROWS_PER_BLOCK * CH + ch;
    const float* p0 = x + base + (long long)(2 * h) * CH;
    const float* p1 = x + base + (long long)(2 * h + 1) * CH;

    v2f a; a.x = 1.0f; a.y = 1.0f;       // all-ones A matrix (layout independent)
    v8f c = {0.f, 0.f, 0.f, 0.f, 0.f, 0.f, 0.f, 0.f};

#pragma unroll 4
    for (int r = 0; r < ROWS_PER_BLOCK; r += 4) {
        float x0 = __builtin_nontemporal_load(p0 + (long long)r * CH);
        float x1 = __builtin_nontemporal_load(p1 + (long long)r * CH);
        v2f b; b.x = x0 * x0; b.y = x1 * x1;
        // D = A(ones) x B(x^2) + C : exact f32 column-sum accumulation
        c = __builtin_amdgcn_wmma_f32_16x16x4_f32(
                /*neg_a=*/false, a, /*neg_b=*/false, b,
                /*c_mod=*/(short)0, c, /*reuse_a=*/false, /*reuse_b=*/false);
    }

    // Every row of D equals the full column sum; lane l (<16) holds channel ch in c[0].
    if (lane < 16) atomicAdd(&e[ch], c[0]);
}

// 96 threads: windowed sum (lo = max(0,i-3), hi = min(95,i+2)) and inverse denom.
__global__ void lrn_denom(const float* __restrict__ e, float* __restrict__ inv) {
    int i = threadIdx.x;
    if (i >= CH) return;
    int lo = i - 3; if (lo < 0) lo = 0;
    int hi = i + 2; if (hi > CH - 1) hi = CH - 1;
    float s = 0.0f;
    for (int j = lo; j <= hi; ++j) s += e[j];
    inv[i] = powf(K_C + ALPHA_C * s, -BETA_C);
}

// Streaming scale: one float4 per thread, b128 nontemporal load/store.
// 96 floats per row = 24 float4 -> channel vector index = i % 24.
__global__ __launch_bounds__(256) void lrn_scale(const float* __restrict__ x,
                                                 const float* __restrict__ inv,
                                                 float* __restrict__ out, int n4) {
    int i = blockIdx.x * 256 + threadIdx.x;
    if (i >= n4) return;
    const v4f* xv = (const v4f*)x;
    const v4f* iv = (const v4f*)inv;   // 24 vectors, stays hot in cache
    v4f xx = __builtin_nontemporal_load(xv + i);
    v4f d  = iv[i % 24];
    v4f o  = xx * d;
    __builtin_nontemporal_store(o, (v4f*)out + i);
}

extern "C" void kernel_launch(void* const* d_in, const int* in_sizes, int n_in,
                              void* d_out, int out_size, void* d_ws, size_t ws_size,
                              hipStream_t stream) {
    const float* x  = (const float*)d_in[0];
    float* out      = (float*)d_out;
    float* e        = (float*)d_ws;   // 96 floats: per-channel energy
    float* inv      = e + CH;         // 96 floats at +384 B (16B aligned)

    lrn_zero  <<<1, CH, 0, stream>>>(e);
    lrn_reduce<<<RBLOCKS, 192, 0, stream>>>(x, e);
    lrn_denom <<<1, CH, 0, stream>>>(e, inv);

    int n4 = in_sizes[0] / 4;   // 25,165,824 float4s
    lrn_scale<<<(n4 + 255) / 256, 256, 0, stream>>>(x, inv, out, n4);
}